// InputLayer_43224550867860
// MI455X (gfx1250) — compile-verified
//
#include <hip/hip_runtime.h>
#include <hip/hip_bf16.h>
#include <math.h>

typedef __attribute__((ext_vector_type(2))) float v2f;
typedef __attribute__((ext_vector_type(8))) float v8f;

#define HALF_V 256
#define NUM_JT (HALF_V / 16)   // 16 column tiles of 16 freqs

// Fourier feature embedding: out[n, 2j] = sin(x[n]*freq[j]), out[n, 2j+1] = cos(...)
// phase matrix is a rank-1 GEMM -> built with V_WMMA_F32_16X16X4_F32 per 16x16 tile.
__global__ __launch_bounds__(256) void fourier_embed_kernel(
    const float* __restrict__ x, float* __restrict__ out,
    int rowTiles, int wavesTotal) {
  const int tid        = threadIdx.x;
  const int lane       = tid & 31;
  const int nlo        = lane & 15;   // N within tile / M-group select
  const int hi         = lane >> 4;   // 0: M=v, 1: M=v+8
  const int waveGlobal = blockIdx.x * (blockDim.x >> 5) + (tid >> 5);

  const int jt       = waveGlobal & (NUM_JT - 1);   // which 16-freq column tile
  int       rt       = waveGlobal >> 4;             // starting row tile
  const int rtStride = wavesTotal >> 4;

  // freq[j] = 10^(j*4/255); evaluate once per wave in f64 so the f32 value is
  // the correctly-rounded one (matches the reference's f32 freqs to ~1 ulp).
  const int    jj      = jt * 16 + nlo;
  const double LOG2_10 = 3.3219280948873623478703194294894;
  const float  freqf   = (float)exp2((double)jj * (4.0 / 255.0) * LOG2_10);

  // B (4x16, K x N): row K=0 carries freqs. Replicate across both lane halves
  // of VGPR0 so whichever half encodes K=0 sees freq[n]; all other K rows of A
  // are zero, so exactly one product term survives: D[m,n] = x[m]*freq[n].
  v2f B; B.x = freqf; B.y = 0.0f;
  const v8f Czero = {};

  // 3-term Cody-Waite split of 2*pi. C1 has 7 mantissa bits; k <= ~8000 (13
  // bits) so k*C1 is exact in f32. Residual error ~1e-6 rad at max phase.
  const double TWO_PI = 6.283185307179586476925286766559;
  const float  C1     = 6.28125f;
  const float  C2     = (float)(TWO_PI - (double)6.28125f);
  const float  C3     = (float)(TWO_PI - (double)6.28125f - (double)(float)(TWO_PI - (double)6.28125f));
  const float  INV2PI = 0.15915494309189535f;   // 1/(2*pi)

  for (; rt < rowTiles; rt += rtStride) {
    const int n0 = rt * 16;

    // A (16x4, M x K): lanes 0-15 VGPR0 = column K=0 = x[n0..n0+15];
    // lanes 16-31 (K=2) and VGPR1 (K=1,3) are zero.
    const float xv = x[n0 + nlo];     // 16 unique addrs, broadcast to both halves
    v2f A; A.x = (hi == 0) ? xv : 0.0f; A.y = 0.0f;

    // 16x16 phase tile: lane l, elem v holds phase[M = v + 8*hi, N = nlo]
    v8f ph = __builtin_amdgcn_wmma_f32_16x16x4_f32(
        false, A, false, B, (short)0, Czero, false, false);

#pragma unroll
    for (int v = 0; v < 8; ++v) {
      const float p = ph[v];
      // reduce to revolutions in [-0.5, 0.5] for the hardware sin/cos,
      // which compute sin(2*pi*s) / cos(2*pi*s).
      const float k = __builtin_rintf(p * INV2PI);
      float r = __builtin_fmaf(k, -C1, p);
      r       = __builtin_fmaf(k, -C2, r);
      r       = __builtin_fmaf(k, -C3, r);
      const float frac = r * INV2PI;

      v2f sc;
      sc.x = __builtin_amdgcn_sinf(frac);   // v_sin_f32
      sc.y = __builtin_amdgcn_cosf(frac);   // v_cos_f32

      const size_t row = (size_t)(n0 + v + 8 * hi);
      const size_t idx = (row << 9) + (size_t)(2 * jj);   // n*512 + 2*j
      // 1 GiB streaming output >> 192 MB L2: non-temporal store
      __builtin_nontemporal_store(sc, (v2f*)(out + idx));
    }
  }
}

extern "C" void kernel_launch(void* const* d_in, const int* in_sizes, int n_in,
                              void* d_out, int out_size, void* d_ws, size_t ws_size,
                              hipStream_t stream) {
  const float* x   = (const float*)d_in[0];
  float*       out = (float*)d_out;

  const int batch    = in_sizes[0];     // 524288
  const int rowTiles = batch / 16;      // 32768 (batch is a multiple of 16)

  const int threads    = 256;           // 8 waves per block
  const int blocks     = 1024;          // 8192 waves total
  const int wavesTotal = blocks * (threads / 32);

  hipLaunchKernelGGL(fourier_embed_kernel, dim3(blocks), dim3(threads), 0, stream,
                     x, out, rowTiles, wavesTotal);
}